// FrequencyAttention_15796889715176
// MI455X (gfx1250) — compile-verified
//
#include <hip/hip_runtime.h>
#include <hip/hip_bf16.h>

// ---------------------------------------------------------------------------
// FrequencyAttention, algebraically refactored.
//
// tok[b,s,:] = LN(x[b,s]*W_in + b_in + pe[s]) is affine in x[b,s] up to two
// per-(b,s) scalars (mean folds in, rstd r multiplies). Everything downstream
// lives in an 8-vector basis {A, B_0..B_5, beta}; the DxD projections reduce
// to one 16x512 @ 512x512 fp32 WMMA GEMM per projection, then per-head 8x8
// Gram tables. Main kernel is ~4 KFLOP/batch of scalar math: memory-bound.
// ---------------------------------------------------------------------------

#define Dm   512
#define Hn   8
#define Sn   6
#define HD   64

typedef float v2f __attribute__((ext_vector_type(2)));
typedef float v8f __attribute__((ext_vector_type(8)));

// workspace float offsets
#define OFF_GQ    0          // 16x512
#define OFF_GK    8192       // 16x512
#define OFF_GV    16384      // 16x512
#define OFF_G     24576      // 16x512 basis
#define OFF_C     32768      // consts block, contiguous:
#define C_GRAM    0          //   8 heads * 64  = 512
#define C_WOA     512        //   8
#define C_WOB     520        //   48
#define C_WOC     568        //   8
#define C_P2      576        //   1
#define C_P1      577        //   6
#define C_P0      583        //   6
#define C_FB      589        //   36
#define C_TOTAL   625

// ---------------------------------------------------------------------------
// Kernel 1: build basis G (rows: A, B_0..B_5, beta, zeros) + LN stat scalars
// + frequency bias table. One block, 512 threads (one per d).
// ---------------------------------------------------------------------------
__global__ __launch_bounds__(512)
void fa_basis_kernel(const float* __restrict__ W_in,
                     const float* __restrict__ b_in,
                     const float* __restrict__ gamma,
                     const float* __restrict__ beta,
                     float* __restrict__ ws) {
    __shared__ float red[512];
    const int d = threadIdx.x;

    const float freqs[Sn] = {0.2f, 0.4f, 0.6f, 0.8f, 1.0f, 0.0f};
    const float w = W_in[d];
    const float g = gamma[d];

    // positional encoding for this dim
    const int   j   = d >> 1;
    const float div = expf(-(float)(2 * j) * (logf(10000.0f) / (float)Dm));
    float c[Sn];
    #pragma unroll
    for (int s = 0; s < Sn; ++s) {
        const float arg = freqs[s] * div;
        const float pe  = (d & 1) ? cosf(arg) : sinf(arg);
        c[s] = b_in[d] + pe;
    }

    // block mean reduction over 512 lanes
    auto rmean = [&](float v) -> float {
        red[d] = v; __syncthreads();
        for (int off = 256; off > 0; off >>= 1) {
            if (d < off) red[d] += red[d + off];
            __syncthreads();
        }
        const float r = red[0] * (1.0f / (float)Dm);
        __syncthreads();
        return r;
    };

    const float mW = rmean(w);
    const float At = w - mW;

    float Bt[Sn];
    #pragma unroll
    for (int s = 0; s < Sn; ++s) {
        const float mC = rmean(c[s]);
        Bt[s] = c[s] - mC;
    }

    const float p2 = rmean(At * At);
    float p1[Sn], p0[Sn];
    #pragma unroll
    for (int s = 0; s < Sn; ++s) p1[s] = rmean(At * Bt[s]);
    #pragma unroll
    for (int s = 0; s < Sn; ++s) p0[s] = rmean(Bt[s] * Bt[s]);

    // write basis rows (16 x 512, rows 8..15 zero so the WMMA M-pad is clean)
    float* G = ws + OFF_G;
    G[0 * Dm + d] = At * g;
    #pragma unroll
    for (int s = 0; s < Sn; ++s) G[(1 + s) * Dm + d] = Bt[s] * g;
    G[7 * Dm + d] = beta[d];
    #pragma unroll
    for (int rrow = 8; rrow < 16; ++rrow) G[rrow * Dm + d] = 0.0f;

    if (d == 0) {
        float* cb = ws + OFF_C;
        cb[C_P2] = p2;
        for (int s = 0; s < Sn; ++s) { cb[C_P1 + s] = p1[s]; cb[C_P0 + s] = p0[s]; }
        // frequency bias
        float fb[Sn * Sn];
        for (int i = 0; i < Sn * Sn; ++i) fb[i] = 0.0f;
        for (int i = 0; i < Sn - 1; ++i) { fb[i * Sn + i + 1] = 0.1f; fb[(i + 1) * Sn + i] = 0.1f; }
        fb[0 * Sn + 1] = 0.05f; fb[1 * Sn + 0] = 0.05f;
        fb[1 * Sn + 3] = 0.05f; fb[3 * Sn + 1] = 0.05f;
        for (int i = 0; i < Sn * Sn; ++i) cb[C_FB + i] = fb[i];
    }
}

// ---------------------------------------------------------------------------
// Kernel 2: GP = G(16x512) @ WP(512x512) (+bias into row 7) in fp32 WMMA.
// 96 waves: 3 projections x 32 column tiles. One wave per 16x16 output tile
// column strip; 128 steps of V_WMMA_F32_16X16X4_F32 along K.
// ISA 7.12.2 layouts: A 16x4 -> lanes0-15 hold K=k0,k0+1; lanes16-31 K=k0+2,+3.
// C/D: VGPR r -> row r (lanes 0-15) and row r+8 (lanes 16-31).
// ---------------------------------------------------------------------------
__global__ __launch_bounds__(32)
void fa_basis_gemm_wmma(const float* __restrict__ Wq, const float* __restrict__ bq,
                        const float* __restrict__ Wk, const float* __restrict__ bk,
                        const float* __restrict__ Wv, const float* __restrict__ bv,
                        float* __restrict__ ws) {
    const int bid  = blockIdx.x;          // 0..95
    const int proj = bid >> 5;            // 0..2
    const int n0   = (bid & 31) << 4;     // column tile base

    const float* W    = (proj == 0) ? Wq : (proj == 1) ? Wk : Wv;
    const float* bias = (proj == 0) ? bq : (proj == 1) ? bk : bv;
    float*       out  = ws + ((proj == 0) ? OFF_GQ : (proj == 1) ? OFF_GK : OFF_GV);
    const float* G    = ws + OFF_G;

    const int lane = threadIdx.x;
    const int m    = lane & 15;
    const int hi   = lane >> 4;

    v8f acc = {};
    for (int k0 = 0; k0 < Dm; k0 += 4) {
        const int kb = k0 + hi * 2;
        v2f a, b;
        a.x = G[m * Dm + kb];
        a.y = G[m * Dm + kb + 1];
        b.x = W[kb * Dm + n0 + m];
        b.y = W[(kb + 1) * Dm + n0 + m];
        acc = __builtin_amdgcn_wmma_f32_16x16x4_f32(false, a, false, b,
                                                    (short)0, acc, false, false);
    }

    const int n = n0 + m;
    #pragma unroll
    for (int vr = 0; vr < 8; ++vr) {
        const int mr  = vr + hi * 8;
        float     val = acc[vr];
        if (mr == 7) val += bias[n];   // cP row carries beta@WP + bias
        out[mr * Dm + n] = val;
    }
}

// ---------------------------------------------------------------------------
// Kernel 3: per-head Gram tables Gram[h][p][q] = <GQ_p, GK_q>_head and
// v.Wo scalars. 8 blocks (heads) x 64 threads.
// ---------------------------------------------------------------------------
__global__ __launch_bounds__(64)
void fa_gram_kernel(const float* __restrict__ Wo, float* __restrict__ ws) {
    const int h  = blockIdx.x;
    const int t  = threadIdx.x;
    const int p  = t >> 3;
    const int q  = t & 7;
    const int d0 = h * HD;

    const float* GQ = ws + OFF_GQ;
    const float* GK = ws + OFF_GK;
    const float* GV = ws + OFF_GV;
    float*       cb = ws + OFF_C;

    float s = 0.0f;
    #pragma unroll 8
    for (int d = 0; d < HD; ++d)
        s += GQ[p * Dm + d0 + d] * GK[q * Dm + d0 + d];
    cb[C_GRAM + h * 64 + p * 8 + q] = s;

    if (q == 0) {
        float dv = 0.0f;
        #pragma unroll 8
        for (int d = 0; d < HD; ++d)
            dv += GV[p * Dm + d0 + d] * Wo[d0 + d];
        if (p == 0)      cb[C_WOA + h] = dv;
        else if (p == 7) cb[C_WOC + h] = dv;
        else             cb[C_WOB + h * Sn + (p - 1)] = dv;
    }
}

// ---------------------------------------------------------------------------
// Kernel 4: main. One thread per batch element; constants staged in LDS.
// ---------------------------------------------------------------------------
__global__ __launch_bounds__(256)
void fa_main_kernel(const float* __restrict__ x,
                    const float* __restrict__ bo,
                    const float* __restrict__ ws,
                    float* __restrict__ out, int B) {
    __shared__ float sC[C_TOTAL];
    const int t = threadIdx.x;
    for (int i = t; i < C_TOTAL; i += 256) sC[i] = ws[OFF_C + i];
    __syncthreads();

    const int b = blockIdx.x * 256 + t;
    if (b >= B) return;

    float xs[Sn], r[Sn], u[Sn];
    #pragma unroll
    for (int s = 0; s < Sn; ++s) xs[s] = x[b * Sn + s];

    const float p2 = sC[C_P2];
    #pragma unroll
    for (int s = 0; s < Sn; ++s) {
        const float var = xs[s] * xs[s] * p2 + 2.0f * xs[s] * sC[C_P1 + s] + sC[C_P0 + s];
        r[s] = rsqrtf(var + 1e-5f);
        u[s] = r[s] * xs[s];
    }

    float outv[Sn];
    float am[Sn * Sn];
    #pragma unroll
    for (int i = 0; i < Sn; ++i) outv[i] = 0.0f;
    #pragma unroll
    for (int i = 0; i < Sn * Sn; ++i) am[i] = 0.0f;

    const float scale = 0.125f;  // 64^-0.5

    for (int h = 0; h < Hn; ++h) {
        const float* M   = &sC[C_GRAM + h * 64];
        const float  woA = sC[C_WOA + h];
        const float  woC = sC[C_WOC + h];

        float sc[Sn * Sn];
        #pragma unroll
        for (int i = 0; i < Sn; ++i) {
            #pragma unroll
            for (int j = 0; j < Sn; ++j) {
                float v = u[i] * (u[j] * M[0 * 8 + 0] + r[j] * M[0 * 8 + 1 + j] + M[0 * 8 + 7])
                        + r[i] * (u[j] * M[(1 + i) * 8 + 0] + r[j] * M[(1 + i) * 8 + 1 + j] + M[(1 + i) * 8 + 7])
                        +        (u[j] * M[7 * 8 + 0] + r[j] * M[7 * 8 + 1 + j] + M[7 * 8 + 7]);
                sc[i * Sn + j] = v * scale + sC[C_FB + i * Sn + j];
            }
        }

        #pragma unroll
        for (int i = 0; i < Sn; ++i) {
            float mx = sc[i * Sn + 0];
            #pragma unroll
            for (int j = 1; j < Sn; ++j) mx = fmaxf(mx, sc[i * Sn + j]);
            float e[Sn], sum = 0.0f;
            #pragma unroll
            for (int j = 0; j < Sn; ++j) { e[j] = expf(sc[i * Sn + j] - mx); sum += e[j]; }
            const float inv = 1.0f / sum;
            float vo = 0.0f;
            #pragma unroll
            for (int j = 0; j < Sn; ++j) {
                const float a = e[j] * inv;
                am[i * Sn + j] += a * 0.125f;  // mean over 8 heads
                vo += a * (u[j] * woA + r[j] * sC[C_WOB + h * Sn + j] + woC);
            }
            outv[i] += vo;
        }
    }

    const float bov = bo[0];
    #pragma unroll
    for (int i = 0; i < Sn; ++i) out[b * Sn + i] = outv[i] + bov + xs[i];

    // attn mean: 36 contiguous floats, 16B-aligned (base = 4*(6B + 36b))
    float4* a4 = reinterpret_cast<float4*>(out + (size_t)B * Sn + (size_t)b * (Sn * Sn));
    #pragma unroll
    for (int k = 0; k < 9; ++k)
        a4[k] = make_float4(am[4 * k + 0], am[4 * k + 1], am[4 * k + 2], am[4 * k + 3]);
}

// ---------------------------------------------------------------------------
extern "C" void kernel_launch(void* const* d_in, const int* in_sizes, int n_in,
                              void* d_out, int out_size, void* d_ws, size_t ws_size,
                              hipStream_t stream) {
    const float* x     = (const float*)d_in[0];
    const float* W_in  = (const float*)d_in[1];
    const float* b_in  = (const float*)d_in[2];
    const float* Wq    = (const float*)d_in[3];
    const float* bq    = (const float*)d_in[4];
    const float* Wk    = (const float*)d_in[5];
    const float* bk    = (const float*)d_in[6];
    const float* Wv    = (const float*)d_in[7];
    const float* bv    = (const float*)d_in[8];
    const float* Wo    = (const float*)d_in[9];
    const float* bo    = (const float*)d_in[10];
    const float* gamma = (const float*)d_in[11];
    const float* beta  = (const float*)d_in[12];

    float* out = (float*)d_out;
    float* ws  = (float*)d_ws;
    const int B = in_sizes[0] / Sn;

    fa_basis_kernel<<<1, 512, 0, stream>>>(W_in, b_in, gamma, beta, ws);
    fa_basis_gemm_wmma<<<96, 32, 0, stream>>>(Wq, bq, Wk, bk, Wv, bv, ws);
    fa_gram_kernel<<<8, 64, 0, stream>>>(Wo, ws);
    fa_main_kernel<<<(B + 255) / 256, 256, 0, stream>>>(x, bo, ws, out, B);
}